// Attention_50534585205236
// MI455X (gfx1250) — compile-verified
//
#include <hip/hip_runtime.h>
#include <hip/hip_bf16.h>
#include <math.h>

// Problem constants (from reference setup_inputs)
#define B_    256
#define L_    196
#define RNN_  1024
#define HID_  512
#define FEAT_ 2048

typedef __attribute__((ext_vector_type(2))) float v2f;
typedef __attribute__((ext_vector_type(8))) float v8f;

// ---------------------------------------------------------------------------
// Kernel 1: Wah = h [B,RNN] x Wah_w^T [RNN,HID]  via V_WMMA_F32_16X16X4_F32.
// One wave32 per 16x16 output tile. 16 M-tiles x 32 N-tiles = 512 tiles.
// 8 waves / block (256 threads) -> 64 blocks.
// A-frag (16x4 f32): lanes 0-15 hold M=lane with K=k,k+1 ; lanes 16-31 hold
// K=k+2,k+3 (ISA 7.12.2). B-frag (4x16) mirrors with N across lanes; since
// B[k][n] = Wah_w[n][k], we read Wah_w rows directly.
// ---------------------------------------------------------------------------
__global__ void wah_gemm_wmma(const float* __restrict__ h,
                              const float* __restrict__ W,
                              float* __restrict__ wah)
{
    const int wave = threadIdx.x >> 5;
    const int lane = threadIdx.x & 31;
    const int tile = blockIdx.x * 8 + wave;      // 0..511
    const int tm   = tile >> 5;                  // 0..15  (M tile)
    const int tn   = tile & 31;                  // 0..31  (N tile)
    const int half = lane >> 4;                  // 0/1 -> K sub-pair
    const int l16  = lane & 15;

    const float* __restrict__ aptr = h + (size_t)(tm * 16 + l16) * RNN_ + half * 2;
    const float* __restrict__ bptr = W + (size_t)(tn * 16 + l16) * RNN_ + half * 2;

    v8f acc0 = {};
    v8f acc1 = {};
    #pragma unroll 4
    for (int k = 0; k < RNN_; k += 8) {
        v2f a0 = *(const v2f*)(aptr + k);
        v2f b0 = *(const v2f*)(bptr + k);
        v2f a1 = *(const v2f*)(aptr + k + 4);
        v2f b1 = *(const v2f*)(bptr + k + 4);
        acc0 = __builtin_amdgcn_wmma_f32_16x16x4_f32(false, a0, false, b0,
                                                     (short)0, acc0, false, false);
        acc1 = __builtin_amdgcn_wmma_f32_16x16x4_f32(false, a1, false, b1,
                                                     (short)0, acc1, false, false);
    }
    v8f acc = acc0 + acc1;

    // C/D layout: VGPR j -> M = tm*16 + j + half*8 ; N = tn*16 + l16
    const int n = tn * 16 + l16;
    #pragma unroll
    for (int j = 0; j < 8; ++j) {
        const int m = tm * 16 + j + half * 8;
        wah[(size_t)m * HID_ + n] = acc[j];
    }
}

// ---------------------------------------------------------------------------
// Kernel 2: scores[b][l] = sum_h tanh(Wah[b][h] + p_att[b][l][h]) * alpha_w[h]
// One wave per (b,l). 8 waves/block -> B*L/8 = 6272 blocks.
// Streams p_att_feats (103 MB) exactly once; Wah row + alpha_w live in cache.
// ---------------------------------------------------------------------------
__global__ void scores_kernel(const float* __restrict__ wah,
                              const float* __restrict__ p_att,
                              const float* __restrict__ alpha_w,
                              float* __restrict__ scores)
{
    const int wave = threadIdx.x >> 5;
    const int lane = threadIdx.x & 31;
    const int idx  = blockIdx.x * 8 + wave;          // 0 .. B*L-1
    if (idx >= B_ * L_) return;
    const int b = idx / L_;

    const float* __restrict__ wrow = wah + (size_t)b * HID_;
    const float* __restrict__ prow = p_att + (size_t)idx * HID_;

    float acc = 0.f;
    #pragma unroll
    for (int i = 0; i < HID_ / 32; ++i) {
        const int hh = lane + i * 32;
        const float e = tanhf(wrow[hh] + prow[hh]);
        acc = fmaf(e, alpha_w[hh], acc);
    }
    #pragma unroll
    for (int off = 16; off > 0; off >>= 1)
        acc += __shfl_down(acc, off, 32);
    if (lane == 0) scores[idx] = acc;
}

// ---------------------------------------------------------------------------
// Kernel 3: softmax over L then att[b][f] = sum_l alpha[l] * att_feats[b][l][f]
// One 256-thread block per batch element b. alpha lives in LDS (broadcast
// reads, no bank conflicts). att_feats (411 MB) streamed once, coalesced.
// ---------------------------------------------------------------------------
__global__ void att_kernel(const float* __restrict__ scores,
                           const float* __restrict__ att_feats,
                           float* __restrict__ out)
{
    __shared__ float salpha[L_];
    __shared__ float red[8];
    __shared__ float bcast;

    const int b    = blockIdx.x;
    const int tid  = threadIdx.x;
    const int lane = tid & 31;
    const int wid  = tid >> 5;

    const float* __restrict__ srow = scores + (size_t)b * L_;
    const float s = (tid < L_) ? srow[tid] : -INFINITY;

    // block max
    float m = s;
    #pragma unroll
    for (int off = 16; off > 0; off >>= 1)
        m = fmaxf(m, __shfl_down(m, off, 32));
    if (lane == 0) red[wid] = m;
    __syncthreads();
    if (tid == 0) {
        float mm = red[0];
        #pragma unroll
        for (int i = 1; i < 8; ++i) mm = fmaxf(mm, red[i]);
        bcast = mm;
    }
    __syncthreads();
    const float bmax = bcast;

    // exp + block sum
    const float e = (tid < L_) ? __expf(s - bmax) : 0.f;
    if (tid < L_) salpha[tid] = e;
    float t = e;
    #pragma unroll
    for (int off = 16; off > 0; off >>= 1)
        t += __shfl_down(t, off, 32);
    if (lane == 0) red[wid] = t;
    __syncthreads();
    if (tid == 0) {
        float ss = red[0];
        #pragma unroll
        for (int i = 1; i < 8; ++i) ss += red[i];
        bcast = 1.f / ss;
    }
    __syncthreads();
    const float inv = bcast;
    if (tid < L_) salpha[tid] = salpha[tid] * inv;
    __syncthreads();

    // weighted sum over L for FEAT outputs
    const float* __restrict__ feats = att_feats + (size_t)b * L_ * FEAT_;
    #pragma unroll
    for (int fc = 0; fc < FEAT_; fc += 256) {
        const float* __restrict__ col = feats + fc + tid;
        float acc = 0.f;
        #pragma unroll 4
        for (int l = 0; l < L_; ++l)
            acc = fmaf(salpha[l], col[(size_t)l * FEAT_], acc);
        out[(size_t)b * FEAT_ + fc + tid] = acc;
    }
}

// ---------------------------------------------------------------------------
extern "C" void kernel_launch(void* const* d_in, const int* in_sizes, int n_in,
                              void* d_out, int out_size, void* d_ws, size_t ws_size,
                              hipStream_t stream)
{
    const float* h         = (const float*)d_in[0];   // [B, RNN]
    const float* att_feats = (const float*)d_in[1];   // [B, L, FEAT]
    const float* p_att     = (const float*)d_in[2];   // [B, L, HID]
    const float* Wah_w     = (const float*)d_in[3];   // [HID, RNN]
    const float* alpha_w   = (const float*)d_in[4];   // [1, HID]
    float* out = (float*)d_out;                       // [B, FEAT]

    float* wah    = (float*)d_ws;                     // [B, HID]   512 KB
    float* scores = wah + (size_t)B_ * HID_;          // [B, L]     200 KB

    // 512 tiles / 8 waves-per-block = 64 blocks
    wah_gemm_wmma<<<64, 256, 0, stream>>>(h, Wah_w, wah);
    // B*L = 50176 waves / 8 per block = 6272 blocks
    scores_kernel<<<(B_ * L_) / 8, 256, 0, stream>>>(wah, p_att, alpha_w, scores);
    // one block per batch element
    att_kernel<<<B_, 256, 0, stream>>>(scores, att_feats, out);
}